// SAGEExpert_70531952935577
// MI455X (gfx1250) — compile-verified
//
#include <hip/hip_runtime.h>

typedef float v2f __attribute__((ext_vector_type(2)));
typedef float v8f __attribute__((ext_vector_type(8)));

#define N_NODES 50000
#define N_EDGES 800000

#define BM 64
#define BN 64
#define BK 32
#define LDA 34  // padded (even!) A-tile stride -> aligned float2 frag reads, no bank conflicts
#define LDK 34  // padded (even!) K-stride for transposed weight tiles

// ---------------------------------------------------------------- utilities

__global__ void zero_f32(float* __restrict__ p, long n) {
  long i = (long)blockIdx.x * blockDim.x + threadIdx.x;
  if (i < n) p[i] = 0.0f;
}

__global__ void degree_kernel(const int* __restrict__ dst,
                              float* __restrict__ deg, int n_edges) {
  int e = blockIdx.x * blockDim.x + threadIdx.x;
  if (e < n_edges) atomicAdd(&deg[dst[e]], 1.0f);
}

__global__ void invdeg_kernel(float* __restrict__ deg, int n) {
  int i = blockIdx.x * blockDim.x + threadIdx.x;
  if (i < n) {
    float d = deg[i];
    deg[i] = d > 0.0f ? 1.0f / fmaxf(d, 1.0f) : 0.0f;
  }
}

// Per-edge gather + scatter-add: block e reads row src[e] (coalesced) and
// atomic-adds into row dst[e]. All rows are L2-resident (<=128MB working set),
// so these f32 atomics land in the L2 atomic units, not HBM.
__global__ __launch_bounds__(256)
void scatter_add_kernel(const float* __restrict__ H,
                        const int* __restrict__ src,
                        const int* __restrict__ dst,
                        float* __restrict__ agg,
                        int dim, int n_edges) {
  int e = blockIdx.x;
  if (e >= n_edges) return;
  long s = src[e];
  long d = dst[e];
  const float* hs = H + s * dim;
  float* ad = agg + d * dim;
  for (int i = threadIdx.x; i < dim; i += blockDim.x)
    atomicAdd(&ad[i], hs[i]);
}

// ---------------------------------------------------------------- fused GEMM

__device__ __forceinline__ float elu_f(float x) {
  return x > 0.0f ? x : (__expf(x) - 1.0f);
}

// Out[n,m] = elu( H[n,:]@Wself[:,m] + (invdeg[n]*AGG[n,:])@Wneigh[:,m] + b[m] )
// Block: 256 threads = 8 waves; block tile 64x64.
// Wave tile: 16 rows x 32 cols = two 16x16 f32-WMMA accumulators sharing the
// same A fragment. Weight tiles stored K-contiguous (transposed) in LDS so
// every fragment is one aligned ds_load_b64 into an even VGPR pair.
template <int HAS_NEIGH>
__global__ __launch_bounds__(256)
void sage_gemm_elu(const float* __restrict__ H,
                   const float* __restrict__ AGG,
                   const float* __restrict__ invdeg,
                   const float* __restrict__ Wself,
                   const float* __restrict__ Wneigh,
                   const float* __restrict__ bias,
                   float* __restrict__ Out,
                   int N, int K, int M) {
  __shared__ float lds_h[BM * LDA];   // A self: [row][k]
  __shared__ float lds_a[BM * LDA];   // A neigh (pre-scaled by invdeg)
  __shared__ float lds_ws[BN * LDK];  // W self, transposed: [col][k]
  __shared__ float lds_wn[BN * LDK];  // W neigh, transposed

  const int tid   = threadIdx.x;
  const int wave  = tid >> 5;
  const int lane  = tid & 31;
  const int lhalf = lane >> 4;   // 0: lanes 0-15, 1: lanes 16-31
  const int l15   = lane & 15;

  const int rowBase = blockIdx.x * BM;
  const int colBase = blockIdx.y * BN;
  const int tRow = (wave & 3) * 16;  // 0..48
  const int tCol = (wave >> 2) * 32; // 0,32

  v8f acc0 = {};  // cols tCol +  0..15
  v8f acc1 = {};  // cols tCol + 16..31

  for (int kc = 0; kc < K; kc += BK) {
    // software prefetch of the next K slab of H (global_prefetch_b8)
    if (kc + BK < K) {
      int pr = rowBase + (tid >> 2);
      if (pr < N) __builtin_prefetch(&H[(long)pr * K + kc + BK + (tid & 3) * 8], 0, 1);
    }

    // stage A (self) tile [BM x BK], coalesced, row-guarded
    for (int i = tid; i < BM * BK; i += 256) {
      int r = i >> 5, c = i & 31;
      int row = rowBase + r;
      lds_h[r * LDA + c] = (row < N) ? H[(long)row * K + kc + c] : 0.0f;
    }
    if (HAS_NEIGH) {
      // fold the 1/deg mean-scaling into the neighbor-A fragment
      for (int i = tid; i < BM * BK; i += 256) {
        int r = i >> 5, c = i & 31;
        int row = rowBase + r;
        lds_a[r * LDA + c] =
            (row < N) ? AGG[(long)row * K + kc + c] * invdeg[row] : 0.0f;
      }
    }
    // stage weight tiles [BK x BN] -> transposed into LDS as [col][k]
    for (int i = tid; i < BK * BN; i += 256) {
      int r = i >> 6, c = i & 63;  // r = k row (coalesced over c)
      lds_ws[c * LDK + r] = Wself[(long)(kc + r) * M + colBase + c];
    }
    if (HAS_NEIGH) {
      for (int i = tid; i < BK * BN; i += 256) {
        int r = i >> 6, c = i & 63;
        lds_wn[c * LDK + r] = Wneigh[(long)(kc + r) * M + colBase + c];
      }
    }
    __syncthreads();

#pragma unroll
    for (int k = 0; k < BK; k += 4) {
      const int ak = k + lhalf * 2;  // lanes 16-31 hold K=k+2,k+3
      // A 16x4: lane = M, VGPR pair = {K=ak, K=ak+1} (aligned b64)
      v2f a  = *(const v2f*)&lds_h[(tRow + l15) * LDA + ak];
      // B 4x16: lane = N, VGPR pair = {K=ak, K=ak+1} (aligned b64, transposed tile)
      v2f b0 = *(const v2f*)&lds_ws[(tCol + l15) * LDK + ak];
      v2f b1 = *(const v2f*)&lds_ws[(tCol + 16 + l15) * LDK + ak];
      acc0 = __builtin_amdgcn_wmma_f32_16x16x4_f32(
          false, a, false, b0, (short)0, acc0, false, false);
      acc1 = __builtin_amdgcn_wmma_f32_16x16x4_f32(
          false, a, false, b1, (short)0, acc1, false, false);
      if (HAS_NEIGH) {
        v2f an  = *(const v2f*)&lds_a[(tRow + l15) * LDA + ak];
        v2f bn0 = *(const v2f*)&lds_wn[(tCol + l15) * LDK + ak];
        v2f bn1 = *(const v2f*)&lds_wn[(tCol + 16 + l15) * LDK + ak];
        acc0 = __builtin_amdgcn_wmma_f32_16x16x4_f32(
            false, an, false, bn0, (short)0, acc0, false, false);
        acc1 = __builtin_amdgcn_wmma_f32_16x16x4_f32(
            false, an, false, bn1, (short)0, acc1, false, false);
      }
    }
    __syncthreads();
  }

  // epilogue: bias + ELU + store
  // C layout: VGPR r -> M = tRow + r (lanes 0-15) / tRow + 8 + r (lanes 16-31)
  const int col0 = colBase + tCol + l15;
  const int col1 = col0 + 16;
  const float bv0 = bias[col0];
  const float bv1 = bias[col1];
#pragma unroll
  for (int r = 0; r < 8; r++) {
    int row = rowBase + tRow + r + lhalf * 8;
    if (row < N) {
      Out[(long)row * M + col0] = elu_f(acc0[r] + bv0);
      Out[(long)row * M + col1] = elu_f(acc1[r] + bv1);
    }
  }
}

// ---------------------------------------------------------------- launcher

extern "C" void kernel_launch(void* const* d_in, const int* in_sizes, int n_in,
                              void* d_out, int out_size, void* d_ws,
                              size_t ws_size, hipStream_t stream) {
  const float* features = (const float*)d_in[0];
  const int*   edge_src = (const int*)d_in[1];
  const int*   edge_dst = (const int*)d_in[2];
  const float* W1s  = (const float*)d_in[3];
  const float* W1n  = (const float*)d_in[4];
  const float* b1   = (const float*)d_in[5];
  const float* W2s  = (const float*)d_in[6];
  const float* W2n  = (const float*)d_in[7];
  const float* b2   = (const float*)d_in[8];
  const float* W3s  = (const float*)d_in[9];
  const float* W3n  = (const float*)d_in[10];
  const float* b3   = (const float*)d_in[11];
  const float* Wres = (const float*)d_in[12];
  const float* bres = (const float*)d_in[13];

  const int IN_DIM = 256, H1 = 640, H2 = 320, OUT_DIM = 128;

  float* out_x   = (float*)d_out;                     // [N,128] (first tuple elem)
  float* out_res = out_x + (size_t)N_NODES * OUT_DIM; // [N,128]

  float* invdeg = (float*)d_ws;                       // N
  float* agg    = invdeg + N_NODES;                   // N*640 (reused per layer)
  float* x1     = agg + (size_t)N_NODES * H1;         // N*640
  float* x2     = x1 + (size_t)N_NODES * H1;          // N*320

  const dim3 blk(256);
  const int gRows = (N_NODES + BM - 1) / BM;

  // degrees -> inv_deg (in place)
  zero_f32<<<(N_NODES + 255) / 256, blk, 0, stream>>>(invdeg, N_NODES);
  degree_kernel<<<(N_EDGES + 255) / 256, blk, 0, stream>>>(edge_dst, invdeg, N_EDGES);
  invdeg_kernel<<<(N_NODES + 255) / 256, blk, 0, stream>>>(invdeg, N_NODES);

  // residual branch: res = elu(features @ Wres + bres)
  sage_gemm_elu<0><<<dim3(gRows, OUT_DIM / BN), blk, 0, stream>>>(
      features, nullptr, nullptr, Wres, nullptr, bres, out_res,
      N_NODES, IN_DIM, OUT_DIM);

  // layer 1
  zero_f32<<<(unsigned)(((long)N_NODES * IN_DIM + 255) / 256), blk, 0, stream>>>(
      agg, (long)N_NODES * IN_DIM);
  scatter_add_kernel<<<N_EDGES, blk, 0, stream>>>(features, edge_src, edge_dst,
                                                  agg, IN_DIM, N_EDGES);
  sage_gemm_elu<1><<<dim3(gRows, H1 / BN), blk, 0, stream>>>(
      features, agg, invdeg, W1s, W1n, b1, x1, N_NODES, IN_DIM, H1);

  // layer 2
  zero_f32<<<(unsigned)(((long)N_NODES * H1 + 255) / 256), blk, 0, stream>>>(
      agg, (long)N_NODES * H1);
  scatter_add_kernel<<<N_EDGES, blk, 0, stream>>>(x1, edge_src, edge_dst,
                                                  agg, H1, N_EDGES);
  sage_gemm_elu<1><<<dim3(gRows, H2 / BN), blk, 0, stream>>>(
      x1, agg, invdeg, W2s, W2n, b2, x2, N_NODES, H1, H2);

  // layer 3 -> first tuple output
  zero_f32<<<(unsigned)(((long)N_NODES * H2 + 255) / 256), blk, 0, stream>>>(
      agg, (long)N_NODES * H2);
  scatter_add_kernel<<<N_EDGES, blk, 0, stream>>>(x2, edge_src, edge_dst,
                                                  agg, H2, N_EDGES);
  sage_gemm_elu<1><<<dim3(gRows, OUT_DIM / BN), blk, 0, stream>>>(
      x2, agg, invdeg, W3s, W3n, b3, out_x, N_NODES, H2, OUT_DIM);
}